// GATLayer_43499428774616
// MI455X (gfx1250) — compile-verified
//
#include <hip/hip_runtime.h>

#define BC 16
#define NN 2048
#define FF 128
#define DD 128
#define NEG_BIG  (-1e16f)
#define MINF     (-3.0e38f)
#define LEAK     0.1f
#define ZPAD     132      // floats per staged LDS row: 528 B, 16B-aligned, bank-spread

typedef __attribute__((ext_vector_type(2))) float v2f;
typedef __attribute__((ext_vector_type(4))) float v4f;
typedef __attribute__((ext_vector_type(8))) float v8f;
typedef __attribute__((ext_vector_type(4))) int   v4i;

typedef __attribute__((address_space(1))) v4i* gv4i_t;   // global int4*
typedef __attribute__((address_space(3))) v4i* lv4i_t;   // LDS int4*

#if defined(__has_builtin)
#if __has_builtin(__builtin_amdgcn_global_load_async_to_lds_b128)
#define USE_ASYNC_LDS 1
#endif
#endif

__device__ __forceinline__ v8f wmma_f32(v2f a, v2f b, v8f c) {
    // D = A(16x4,f32) * B(4x16,f32) + C(16x16,f32)
    return __builtin_amdgcn_wmma_f32_16x16x4_f32(
        false, a, false, b, (short)0, c, false, false);
}

__device__ __forceinline__ void wait_async0() {
#if defined(USE_ASYNC_LDS)
#if __has_builtin(__builtin_amdgcn_s_wait_asynccnt)
    __builtin_amdgcn_s_wait_asynccnt(0);
#else
    asm volatile("s_wait_asynccnt 0x0" ::: "memory");
#endif
#endif
}

// ---------------------------------------------------------------------------
// Kernel 1: z = h @ W   ([BC*N,128] x [128,128] -> [BC*N,128]), f32 WMMA
// ---------------------------------------------------------------------------
__global__ __launch_bounds__(256) void zproj_kernel(const float* __restrict__ h,
                                                    const float* __restrict__ W,
                                                    float* __restrict__ z) {
    const int lane = threadIdx.x & 31;
    const int wv   = threadIdx.x >> 5;
    const int lm   = lane & 15;
    const int half = lane >> 4;

    const int tile = blockIdx.x * 8 + wv;       // 16384 tiles total
    const int tm   = tile >> 3;                 // row tile (BC*N/16)
    const int tn   = tile & 7;                  // col tile (D/16)
    const int row  = tm * 16 + lm;
    const int col  = tn * 16 + lm;

    const float* hrow = h + (size_t)row * FF;

    v8f acc = {};
#pragma unroll
    for (int kc = 0; kc < FF / 4; ++kc) {
        const int k = kc * 4 + 2 * half;
        v2f a, b;
        a.x = hrow[k];
        a.y = hrow[k + 1];
        b.x = W[k * DD + col];
        b.y = W[(k + 1) * DD + col];
        acc = wmma_f32(a, b, acc);
    }

    float* zt = z + (size_t)tm * 16 * DD + tn * 16;
#pragma unroll
    for (int r = 0; r < 8; ++r) {
        const int m = half ? r + 8 : r;
        zt[m * DD + lm] = acc[r];
    }
}

// ---------------------------------------------------------------------------
// Kernel 2: scatter COO edges into an N x N bitmask (row-major, 64 words/row)
// ---------------------------------------------------------------------------
__global__ void edge_mask_kernel(const int* __restrict__ row,
                                 const int* __restrict__ col,
                                 unsigned* __restrict__ mask, int nE) {
    const int i = blockIdx.x * blockDim.x + threadIdx.x;
    if (i < nE) {
        const int r = row[i];
        const int c = col[i];
        atomicOr(&mask[r * (NN / 32) + (c >> 5)], 1u << (c & 31));
    }
}

// ---------------------------------------------------------------------------
// Stage one 16-node key tile (16 x 128 f32 = 8 KB) into LDS.
// 512 x 16B segments; thread t moves segments t and t+256 (coalesced).
// ---------------------------------------------------------------------------
__device__ __forceinline__ void stage_tile_async(const float* __restrict__ zb, int ct,
                                                 float* buf, int t) {
#if defined(USE_ASYNC_LDS)
#pragma unroll
    for (int i = 0; i < 2; ++i) {
        const int seg  = i * 256 + t;
        const int node = seg >> 5;
        const int ks   = (seg & 31) << 2;
        const float* g = zb + (size_t)(ct * 16 + node) * DD + ks;
        float* l = buf + node * ZPAD + ks;
        __builtin_amdgcn_global_load_async_to_lds_b128(
            (gv4i_t)g, (lv4i_t)l, 0, 0);
    }
#endif
}

__device__ __forceinline__ void stage_tile_sync(const float* __restrict__ zb, int ct,
                                                float* buf, int t) {
#pragma unroll
    for (int i = 0; i < 2; ++i) {
        const int seg  = i * 256 + t;
        const int node = seg >> 5;
        const int ks   = (seg & 31) << 2;
        const v4f v = *(const v4f*)(zb + (size_t)(ct * 16 + node) * DD + ks);
        *(v4f*)(buf + node * ZPAD + ks) = v;
    }
}

// ---------------------------------------------------------------------------
// Kernel 3: fused flash-style GAT attention.
// grid = (BC, N/128); 256 threads = 8 waves; wave w owns 16 query rows.
// Key tiles double-buffered in LDS via async global->LDS DMA.
// ---------------------------------------------------------------------------
__global__ __launch_bounds__(256) void gat_attn_kernel(const float* __restrict__ z,
                                                       const unsigned* __restrict__ mask,
                                                       float* __restrict__ out) {
    __shared__ float lds_zc[2][16 * ZPAD];  // double-buffered key tiles
    __shared__ float lds_p[8][16 * 17];     // per-wave P tile, pad 17

    const int b    = blockIdx.x;
    const int rb   = blockIdx.y;
    const int t    = threadIdx.x;
    const int lane = t & 31;
    const int wv   = t >> 5;
    const int lm   = lane & 15;
    const int half = lane >> 4;

    const int rowbase = rb * 128 + wv * 16;      // node index within batch
    const float* zb   = z + (size_t)b * NN * DD;

    // A fragments of this wave's 16 query rows (register resident: 64 VGPRs)
    v2f afrag[FF / 4];
    {
        const float* zr = zb + (size_t)(rowbase + lm) * DD;
#pragma unroll
        for (int kc = 0; kc < FF / 4; ++kc) {
            const int k = kc * 4 + 2 * half;
            afrag[kc].x = zr[k];
            afrag[kc].y = zr[k + 1];
        }
    }

    const v8f vzero = {};
    v8f o[8];
#pragma unroll
    for (int nt = 0; nt < 8; ++nt) o[nt] = vzero;
    float mrow[8], lrow[8];
#pragma unroll
    for (int r = 0; r < 8; ++r) { mrow[r] = MINF; lrow[r] = 0.f; }

    const int NT = NN / 16;
#if defined(USE_ASYNC_LDS)
    stage_tile_async(zb, 0, &lds_zc[0][0], t);   // prefetch tile 0
#endif

    for (int ct = 0; ct < NT; ++ct) {
        float* cur = &lds_zc[ct & 1][0];
#if defined(USE_ASYNC_LDS)
        wait_async0();                 // tile ct resident (this wave's portion)
        __syncthreads();               // all waves' portions resident; prev reads done
        if (ct + 1 < NT)               // overlap next tile's DMA with compute
            stage_tile_async(zb, ct + 1, &lds_zc[(ct + 1) & 1][0], t);
#else
        __syncthreads();
        stage_tile_sync(zb, ct, cur, t);
        __syncthreads();
#endif

        // S = z_rows @ z_cols^T  (16x16, f32 WMMA over K=128)
        v8f s = vzero;
#pragma unroll
        for (int kc = 0; kc < FF / 4; ++kc) {
            const int k = kc * 4 + 2 * half;
            v2f bf;
            bf.x = cur[lm * ZPAD + k];
            bf.y = cur[lm * ZPAD + k + 1];
            s = wmma_f32(afrag[kc], bf, s);
        }

        // leaky-relu + adjacency mask + online softmax
        float p[8], scale[8];
#pragma unroll
        for (int r = 0; r < 8; ++r) {
            const int m    = half ? r + 8 : r;
            const int grow = rowbase + m;
            const unsigned mw  = mask[grow * (NN / 32) + (ct >> 1)];
            const unsigned bit = (mw >> (((ct & 1) << 4) + lm)) & 1u;
            const float v  = s[r];
            const float lr = v > 0.f ? v : LEAK * v;
            p[r] = (bit && v != 0.f) ? lr : NEG_BIG;

            float tmx = p[r];                       // row max over 16 key lanes
            for (int off = 1; off < 16; off <<= 1)
                tmx = fmaxf(tmx, __shfl_xor(tmx, off, 16));

            const float mnew = fmaxf(mrow[r], tmx);
            scale[r] = __expf(mrow[r] - mnew);
            mrow[r]  = mnew;

            const float e = __expf(p[r] - mnew);
            float esum = e;
            for (int off = 1; off < 16; off <<= 1)
                esum += __shfl_xor(esum, off, 16);
            lrow[r] = lrow[r] * scale[r] + esum;
            p[r] = e;                               // probability weight
        }

        // C-layout -> A-layout: park P tile in padded LDS (same-wave, DS in-order)
#pragma unroll
        for (int r = 0; r < 8; ++r) {
            const int m = half ? r + 8 : r;
            lds_p[wv][m * 17 + lm] = p[r];
        }

        // rescale accumulators by per-row exp(m_old - m_new)
#pragma unroll
        for (int nt = 0; nt < 8; ++nt)
#pragma unroll
            for (int r = 0; r < 8; ++r)
                o[nt][r] *= scale[r];

        // O += P(16x16) @ z_cols(16x128), f32 WMMA, K = 16 in 4 chunks
#pragma unroll
        for (int kc = 0; kc < 4; ++kc) {
            const int k = kc * 4 + 2 * half;
            v2f a;
            a.x = lds_p[wv][lm * 17 + k];
            a.y = lds_p[wv][lm * 17 + k + 1];
#pragma unroll
            for (int nt = 0; nt < 8; ++nt) {
                v2f bf;
                bf.x = cur[k * ZPAD + nt * 16 + lm];
                bf.y = cur[(k + 1) * ZPAD + nt * 16 + lm];
                o[nt] = wmma_f32(a, bf, o[nt]);
            }
        }
    }

    // normalize and write out (coalesced 64B per lane-half per row)
    float inv[8];
#pragma unroll
    for (int r = 0; r < 8; ++r) inv[r] = 1.f / lrow[r];
    float* ob = out + ((size_t)b * NN + rowbase) * DD;
#pragma unroll
    for (int nt = 0; nt < 8; ++nt)
#pragma unroll
        for (int r = 0; r < 8; ++r) {
            const int m = half ? r + 8 : r;
            ob[m * DD + nt * 16 + lm] = o[nt][r] * inv[r];
        }
}

// ---------------------------------------------------------------------------
extern "C" void kernel_launch(void* const* d_in, const int* in_sizes, int n_in,
                              void* d_out, int out_size, void* d_ws, size_t ws_size,
                              hipStream_t stream) {
    const float* h   = (const float*)d_in[0];
    const float* W   = (const float*)d_in[1];
    const int*   row = (const int*)d_in[2];
    const int*   col = (const int*)d_in[3];
    float* out = (float*)d_out;

    float*    z    = (float*)d_ws;                                   // 16 MB
    unsigned* mask = (unsigned*)((char*)d_ws +
                                 (size_t)BC * NN * DD * sizeof(float));
    const int nE = in_sizes[2];

    (void)hipMemsetAsync(mask, 0, (size_t)NN * (NN / 32) * sizeof(unsigned), stream);
    edge_mask_kernel<<<(nE + 255) / 256, 256, 0, stream>>>(row, col, mask, nE);
    zproj_kernel<<<BC * NN / 16, 256, 0, stream>>>(h, W, z);
    dim3 g(BC, NN / 128);
    gat_attn_kernel<<<g, 256, 0, stream>>>(z, mask, out);
}